// MLP2_71047349011134
// MI455X (gfx1250) — compile-verified
//
#include <hip/hip_runtime.h>

// MLP2: out = (silu(LN(silu(LN(x@W1.T+b1))@W2.T+b2)))@W3.T + b3
// N = 2,097,152 rows, D=1, H=32, all f32.
//
// One wave32 handles a 16-row tile. Stage-1 activations are produced directly
// in V_WMMA_F32_16X16X4_F32 A-fragment layout; the 16x32 @ 32x32 second-layer
// GEMM is 16 f32 WMMAs (8 K-steps x 2 N-tiles); LayerNorm / SiLU / final dot
// are half-wave shuffle reductions on the D-fragment layout.

typedef __attribute__((ext_vector_type(2))) float v2f;
typedef __attribute__((ext_vector_type(8))) float v8f;

#define LN_EPS 1e-5f

__device__ __forceinline__ float silu_f(float v) {
    // v * sigmoid(v) = v / (1 + exp(-v)); exp via exp2, divide via v_rcp_f32
    float e = __builtin_amdgcn_exp2f(-1.44269504088896341f * v);
    return v * __builtin_amdgcn_rcpf(1.0f + e);
}

// sum across the 16 lanes of a half-wave (masks < 16 never cross halves)
__device__ __forceinline__ float half_reduce_add(float v) {
    v += __shfl_xor(v, 1, 32);
    v += __shfl_xor(v, 2, 32);
    v += __shfl_xor(v, 4, 32);
    v += __shfl_xor(v, 8, 32);
    return v;
}

__global__ __launch_bounds__(256) void mlp2_wmma_kernel(
    const float* __restrict__ x,
    const float* __restrict__ W1, const float* __restrict__ b1,
    const float* __restrict__ g1, const float* __restrict__ be1,
    const float* __restrict__ W2, const float* __restrict__ b2,
    const float* __restrict__ g2, const float* __restrict__ be2,
    const float* __restrict__ W3, const float* __restrict__ b3,
    float* __restrict__ out, int ntiles)
{
    const int lane = threadIdx.x & 31;
    const int l16  = lane & 15;   // row within tile (A), or column n (B/D)
    const int h    = lane >> 4;   // half-wave select: K offset 2h in each 4-group

    // ---- loop-invariant per-lane constants (registers) ----
    // Stage-1 params at this lane's fixed hidden indices j = 4k + 2h (+1)
    float w1a[8], w1b[8], b1a[8], b1b[8], g1a[8], g1b[8], e1a[8], e1b[8];
#pragma unroll
    for (int k = 0; k < 8; ++k) {
        int j0 = 4 * k + 2 * h;
        w1a[k] = W1[j0];  w1b[k] = W1[j0 + 1];
        b1a[k] = b1[j0];  b1b[k] = b1[j0 + 1];
        g1a[k] = g1[j0];  g1b[k] = g1[j0 + 1];
        e1a[k] = be1[j0]; e1b[k] = be1[j0 + 1];
    }
    // B fragments of W2^T (B[k][n] = W2[n*32+k]); VGPR0 = K even-of-pair, VGPR1 = +1
    float B0a[8], B0b[8], B1a[8], B1b[8];
#pragma unroll
    for (int k = 0; k < 8; ++k) {
        int kk = 4 * k + 2 * h;
        B0a[k] = W2[(l16     ) * 32 + kk];
        B0b[k] = W2[(l16     ) * 32 + kk + 1];
        B1a[k] = W2[(l16 + 16) * 32 + kk];
        B1b[k] = W2[(l16 + 16) * 32 + kk + 1];
    }
    const float b2lo = b2[l16],  b2hi = b2[l16 + 16];
    const float g2lo = g2[l16],  g2hi = g2[l16 + 16];
    const float e2lo = be2[l16], e2hi = be2[l16 + 16];
    const float w3lo = W3[l16],  w3hi = W3[l16 + 16];
    const float b3v  = b3[0];

    const int wave   = (int)((blockIdx.x * blockDim.x + threadIdx.x) >> 5);
    const int nwaves = (int)((gridDim.x * blockDim.x) >> 5);

    // Wave-uniform grid-stride loop -> EXEC stays all-1s at every WMMA.
    for (int tile = wave; tile < ntiles; tile += nwaves) {
        const int base = tile << 4;
        const float xv = x[base + l16];   // lanes L and L+16 read the same row

        // ---- stage 1: h1 = silu(LN(x*W1 + b1)), built in A-fragment layout ----
        float a0[8], a1[8];
        float s = 0.f, sq = 0.f;
#pragma unroll
        for (int k = 0; k < 8; ++k) {
            float p0 = fmaf(xv, w1a[k], b1a[k]);
            float p1 = fmaf(xv, w1b[k], b1b[k]);
            a0[k] = p0; a1[k] = p1;
            s += p0 + p1;
            sq = fmaf(p0, p0, fmaf(p1, p1, sq));
        }
        // row's other 16 values live in the partner lane (xor 16)
        s  += __shfl_xor(s, 16, 32);
        sq += __shfl_xor(sq, 16, 32);
        const float mu  = s * (1.0f / 32.0f);
        const float var = sq * (1.0f / 32.0f) - mu * mu;
        const float rin = __builtin_amdgcn_rsqf(var + LN_EPS);
#pragma unroll
        for (int k = 0; k < 8; ++k) {
            a0[k] = silu_f(fmaf((a0[k] - mu) * rin, g1a[k], e1a[k]));
            a1[k] = silu_f(fmaf((a1[k] - mu) * rin, g1b[k], e1b[k]));
        }

        // ---- stage 2 GEMM: [16x32] @ [32x32] = 16 x V_WMMA_F32_16X16X4_F32 ----
        v8f acc0 = {};  // output columns n = 0..15
        v8f acc1 = {};  // output columns n = 16..31
#pragma unroll
        for (int k = 0; k < 8; ++k) {
            v2f A;   A.x   = a0[k];  A.y   = a1[k];
            v2f Bt0; Bt0.x = B0a[k]; Bt0.y = B0b[k];
            v2f Bt1; Bt1.x = B1a[k]; Bt1.y = B1b[k];
            acc0 = __builtin_amdgcn_wmma_f32_16x16x4_f32(
                false, A, false, Bt0, (short)0, acc0, false, false);
            acc1 = __builtin_amdgcn_wmma_f32_16x16x4_f32(
                false, A, false, Bt1, (short)0, acc1, false, false);
        }

        // ---- bias + LN + SiLU + W3 dot, per row r of the D fragments ----
        // D layout: VGPR r, lanes 0-15 -> row base+r ; lanes 16-31 -> row base+8+r
#pragma unroll
        for (int r = 0; r < 8; ++r) {
            float v0 = acc0[r] + b2lo;
            float v1 = acc1[r] + b2hi;
            float s2 = half_reduce_add(v0 + v1);
            float q2 = half_reduce_add(fmaf(v0, v0, v1 * v1));
            float mu2  = s2 * (1.0f / 32.0f);
            float var2 = q2 * (1.0f / 32.0f) - mu2 * mu2;
            float ri2  = __builtin_amdgcn_rsqf(var2 + LN_EPS);
            v0 = silu_f(fmaf((v0 - mu2) * ri2, g2lo, e2lo));
            v1 = silu_f(fmaf((v1 - mu2) * ri2, g2hi, e2hi));
            float t = half_reduce_add(fmaf(v0, w3lo, v1 * w3hi));
            if (l16 == 0) out[base + h * 8 + r] = t + b3v;  // lanes 0 and 16 write
        }
    }
}

extern "C" void kernel_launch(void* const* d_in, const int* in_sizes, int n_in,
                              void* d_out, int out_size, void* d_ws, size_t ws_size,
                              hipStream_t stream) {
    const float* x   = (const float*)d_in[0];
    const float* W1  = (const float*)d_in[1];
    const float* b1  = (const float*)d_in[2];
    const float* g1  = (const float*)d_in[3];
    const float* be1 = (const float*)d_in[4];
    const float* W2  = (const float*)d_in[5];
    const float* b2  = (const float*)d_in[6];
    const float* g2  = (const float*)d_in[7];
    const float* be2 = (const float*)d_in[8];
    const float* W3  = (const float*)d_in[9];
    const float* b3  = (const float*)d_in[10];
    float* out = (float*)d_out;

    const int n = in_sizes[0];      // N rows (D == 1); N = 2^21 is a multiple of 16
    const int ntiles = n >> 4;

    const int threads = 256;        // 8 waves per block
    const int blocks  = 1024;       // 8192 waves -> 16 tiles per wave, grid-stride

    mlp2_wmma_kernel<<<blocks, threads, 0, stream>>>(
        x, W1, b1, g1, be1, W2, b2, g2, be2, W3, b3, out, ntiles);
}